// AttentionSuper_33346126086550
// MI455X (gfx1250) — compile-verified
//
#include <hip/hip_runtime.h>

// ============================================================================
// AutoFormer super-attention with 2D relative-position bias, CDNA5 (gfx1250).
//
//   attn = softmax( (Q·Kᵀ + Q·rel_kᵀ) * D^-1/2 )
//   out  = attn·V + attn·rel_v      (laid out [B, N, H*D])
//
// rel_k/rel_v are never materialized: since rel_k[i,j] = Tkv[iv]+Tkh[ih] with
// only 30 distinct table rows, Q·rel_kᵀ collapses to P = Q·Tᵀ (WMMA) + gather,
// and attn·rel_v collapses to W·T (WMMA) where W[i,r] = Σ_{j:idx=r} attn[i,j]
// is built with LDS float atomics.  All GEMMs use V_WMMA_F32_16X16X4_F32
// (exact fp32 path; wave32 fragment layouts per ISA §7.12.2).
// ============================================================================

typedef __attribute__((ext_vector_type(2))) float v2f;
typedef __attribute__((ext_vector_type(8))) float v8f;

#define B_    64
#define N_    197
#define H_    10
#define D_    64
#define C3_   (3 * H_ * D_)   // 1920 floats per token row of x
#define NT_   13              // ceil(197/16) query/key tiles
#define NPAD  208             // 13*16, zero-padded key/value rows
#define TROWS 30              // relative-position table rows
#define TPAD  32              // padded to 2 WMMA tiles

// ---- LDS layout (float offsets) --------------------------------------------
#define OFF_K   0                       // K  [NPAD x 64]
#define OFF_V   (OFF_K + NPAD * D_)     // V  [NPAD x 64]
#define OFF_QW  (OFF_V + NPAD * D_)     // Q  [16 x 64]  (aliased by Wv/Wh later)
#define OFF_S   (OFF_QW + 16 * D_)      // S/attn [16 x NPAD]
#define OFF_TKV (OFF_S + 16 * NPAD)     // k_table_v [TPAD x 64]
#define OFF_TKH (OFF_TKV + TPAD * D_)
#define OFF_TVV (OFF_TKH + TPAD * D_)
#define OFF_TVH (OFF_TVV + TPAD * D_)
#define OFF_PV  (OFF_TVH + TPAD * D_)   // Pv [16 x TPAD]
#define OFF_PH  (OFF_PV + 16 * TPAD)    // Ph [16 x TPAD]
#define LDS_FLOATS (OFF_PH + 16 * TPAD) // = 40192 floats = 160768 B (< 320 KB WGP LDS)

__device__ __forceinline__ void rel_idx(int gi, int j, int giv, int gih,
                                        int& iv, int& ih) {
    // reference: dist = j//14 - i//14 (and %14), clipped to [-14,14], +15;
    // cls row/col (i==0 or j==0) -> index 0
    if (gi != 0 && j != 0) {
        int dv = j / 14 - giv;
        int dh = j % 14 - gih;
        dv = dv < -14 ? -14 : (dv > 14 ? 14 : dv);
        dh = dh < -14 ? -14 : (dh > 14 ? 14 : dh);
        iv = dv + 15;
        ih = dh + 15;
    } else {
        iv = 0;
        ih = 0;
    }
}

__global__ __launch_bounds__(128)
void attn_super_kernel(const float* __restrict__ x,
                       const float* __restrict__ ktv,
                       const float* __restrict__ kth,
                       const float* __restrict__ vtv,
                       const float* __restrict__ vth,
                       float* __restrict__ out) {
    extern __shared__ float lds[];
    float* sK   = lds + OFF_K;
    float* sV   = lds + OFF_V;
    float* sQ   = lds + OFF_QW;
    float* sWv  = lds + OFF_QW;          // alias: Q dead after S/P GEMMs
    float* sWh  = lds + OFF_QW + 16 * TPAD;
    float* sS   = lds + OFF_S;
    float* sTkv = lds + OFF_TKV;
    float* sTkh = lds + OFF_TKH;
    float* sTvv = lds + OFF_TVV;
    float* sTvh = lds + OFF_TVH;
    float* sPv  = lds + OFF_PV;
    float* sPh  = lds + OFF_PH;

    const int it   = blockIdx.x;          // query tile 0..12
    const int h    = blockIdx.y;
    const int b    = blockIdx.z;
    const int i0   = it * 16;
    const int tid  = threadIdx.x;
    const int lane = tid & 31;
    const int wave = tid >> 5;            // 4 waves
    const int mrow  = lane & 15;          // M (or N/B-row) index within tile
    const int khalf = (lane >> 4) << 1;   // 0 (lanes 0-15) or 2 (lanes 16-31)
    const int mhi   = (lane >> 4) << 3;   // +8 row offset for C/D frag

    const float4 z4 = make_float4(0.f, 0.f, 0.f, 0.f);
    const float* xb = x + (size_t)b * N_ * C3_;

    // ---- stage K, V (padded to 208 rows), Q tile, tables into LDS ----------
    for (int idx = tid; idx < NPAD * 16; idx += 128) {
        int row = idx >> 4, c4 = (idx & 15) << 2;
        float4 kq = z4, vq = z4;
        if (row < N_) {
            const float* p = xb + (size_t)row * C3_ + h * D_ + c4;
            kq = *(const float4*)(p + H_ * D_);       // k slab
            vq = *(const float4*)(p + 2 * H_ * D_);   // v slab
        }
        *(float4*)&sK[row * D_ + c4] = kq;
        *(float4*)&sV[row * D_ + c4] = vq;
    }
    for (int idx = tid; idx < 16 * 16; idx += 128) {
        int row = idx >> 4, c4 = (idx & 15) << 2;
        int gi = i0 + row;
        float4 q = (gi < N_) ? *(const float4*)(xb + (size_t)gi * C3_ + h * D_ + c4) : z4;
        *(float4*)&sQ[row * D_ + c4] = q;
    }
    for (int idx = tid; idx < TPAD * 16; idx += 128) {
        int row = idx >> 4, c4 = (idx & 15) << 2;
        float4 a = z4, bq = z4, c = z4, d = z4;
        if (row < TROWS) {
            a  = *(const float4*)(ktv + row * D_ + c4);
            bq = *(const float4*)(kth + row * D_ + c4);
            c  = *(const float4*)(vtv + row * D_ + c4);
            d  = *(const float4*)(vth + row * D_ + c4);
        }
        *(float4*)&sTkv[row * D_ + c4] = a;
        *(float4*)&sTkh[row * D_ + c4] = bq;
        *(float4*)&sTvv[row * D_ + c4] = c;
        *(float4*)&sTvh[row * D_ + c4] = d;
    }
    __syncthreads();

    // ---- S = Q·Kᵀ : 13 column tiles distributed round-robin over waves -----
    for (int jt = wave; jt < NT_; jt += 4) {
        const int j0 = jt * 16;
        v8f acc = {};
#pragma unroll
        for (int kk = 0; kk < D_ / 4; ++kk) {
            const int kb = kk * 4 + khalf;
            v2f a, bf;
            a.x  = sQ[mrow * D_ + kb];
            a.y  = sQ[mrow * D_ + kb + 1];
            bf.x = sK[(j0 + mrow) * D_ + kb];      // B = Kᵀ, fetched row-major
            bf.y = sK[(j0 + mrow) * D_ + kb + 1];
            acc = __builtin_amdgcn_wmma_f32_16x16x4_f32(false, a, false, bf,
                                                        (short)0, acc, false, false);
        }
#pragma unroll
        for (int r = 0; r < 8; ++r)
            sS[(r + mhi) * NPAD + j0 + mrow] = acc[r];
    }

    // ---- P = Q·Tᵀ : 4 tiles, one per wave ([16 x 30] projections) ----------
    {
        const float* T = (wave < 2) ? sTkv : sTkh;
        float*       P = (wave < 2) ? sPv : sPh;
        const int r0 = (wave & 1) * 16;
        v8f acc = {};
#pragma unroll
        for (int kk = 0; kk < D_ / 4; ++kk) {
            const int kb = kk * 4 + khalf;
            v2f a, bf;
            a.x  = sQ[mrow * D_ + kb];
            a.y  = sQ[mrow * D_ + kb + 1];
            bf.x = T[(r0 + mrow) * D_ + kb];
            bf.y = T[(r0 + mrow) * D_ + kb + 1];
            acc = __builtin_amdgcn_wmma_f32_16x16x4_f32(false, a, false, bf,
                                                        (short)0, acc, false, false);
        }
#pragma unroll
        for (int r = 0; r < 8; ++r)
            P[(r + mhi) * TPAD + r0 + mrow] = acc[r];
    }
    __syncthreads();

    // ---- zero W histograms (alias over dead Q) -----------------------------
    for (int idx = tid; idx < 2 * 16 * TPAD; idx += 128) sWv[idx] = 0.0f;
    __syncthreads();

    // ---- bias + scale + softmax + W scatter (8 lanes per query row) --------
    {
        const int row = tid >> 3;          // 0..15
        const int lj  = tid & 7;
        const int gi  = i0 + row;
        const int giv = gi / 14, gih = gi % 14;
        const float scale = 0.125f;        // D^-1/2

        float mx = -3.0e38f;
        for (int j = lj; j < N_; j += 8) {
            int iv, ih;
            rel_idx(gi, j, giv, gih, iv, ih);
            float s = (sS[row * NPAD + j] + sPv[row * TPAD + iv]
                                          + sPh[row * TPAD + ih]) * scale;
            sS[row * NPAD + j] = s;
            mx = fmaxf(mx, s);
        }
        mx = fmaxf(mx, __shfl_xor(mx, 1));
        mx = fmaxf(mx, __shfl_xor(mx, 2));
        mx = fmaxf(mx, __shfl_xor(mx, 4));

        float sum = 0.f;
        for (int j = lj; j < N_; j += 8) {
            float e = __expf(sS[row * NPAD + j] - mx);
            sS[row * NPAD + j] = e;
            sum += e;
        }
        sum += __shfl_xor(sum, 1);
        sum += __shfl_xor(sum, 2);
        sum += __shfl_xor(sum, 4);
        const float inv = 1.0f / sum;

        for (int j = lj; j < NPAD; j += 8) {
            if (j < N_) {
                float a = sS[row * NPAD + j] * inv;
                sS[row * NPAD + j] = a;
                int iv, ih;
                rel_idx(gi, j, giv, gih, iv, ih);
                atomicAdd(&sWv[row * TPAD + iv], a);   // ds_add_f32
                atomicAdd(&sWh[row * TPAD + ih], a);
            } else {
                sS[row * NPAD + j] = 0.0f;             // zero-pad attn for GEMM
            }
        }
    }
    __syncthreads();

    // ---- O = attn·V + Wv·Tvv + Wh·Tvh : one 16-wide d-tile per wave --------
    {
        const int d0 = wave * 16;
        v8f acc = {};
        for (int kk = 0; kk < NPAD / 4; ++kk) {        // content path, K = 208
            const int kb = kk * 4 + khalf;
            v2f a, bf;
            a.x  = sS[mrow * NPAD + kb];
            a.y  = sS[mrow * NPAD + kb + 1];
            bf.x = sV[kb * D_ + d0 + mrow];
            bf.y = sV[(kb + 1) * D_ + d0 + mrow];
            acc = __builtin_amdgcn_wmma_f32_16x16x4_f32(false, a, false, bf,
                                                        (short)0, acc, false, false);
        }
#pragma unroll
        for (int kk = 0; kk < TPAD / 4; ++kk) {        // relative-value path, K = 32
            const int kb = kk * 4 + khalf;
            v2f a, bf;
            a.x  = sWv[mrow * TPAD + kb];
            a.y  = sWv[mrow * TPAD + kb + 1];
            bf.x = sTvv[kb * D_ + d0 + mrow];
            bf.y = sTvv[(kb + 1) * D_ + d0 + mrow];
            acc = __builtin_amdgcn_wmma_f32_16x16x4_f32(false, a, false, bf,
                                                        (short)0, acc, false, false);
            a.x  = sWh[mrow * TPAD + kb];
            a.y  = sWh[mrow * TPAD + kb + 1];
            bf.x = sTvh[kb * D_ + d0 + mrow];
            bf.y = sTvh[(kb + 1) * D_ + d0 + mrow];
            acc = __builtin_amdgcn_wmma_f32_16x16x4_f32(false, a, false, bf,
                                                        (short)0, acc, false, false);
        }
#pragma unroll
        for (int r = 0; r < 8; ++r) {
            const int gi = i0 + r + mhi;
            if (gi < N_)
                out[((size_t)(b * N_ + gi)) * (H_ * D_) + h * D_ + d0 + mrow] = acc[r];
        }
    }
}

extern "C" void kernel_launch(void* const* d_in, const int* in_sizes, int n_in,
                              void* d_out, int out_size, void* d_ws, size_t ws_size,
                              hipStream_t stream) {
    const float* x   = (const float*)d_in[0];
    const float* ktv = (const float*)d_in[1];
    const float* kth = (const float*)d_in[2];
    const float* vtv = (const float*)d_in[3];
    const float* vth = (const float*)d_in[4];
    float* out = (float*)d_out;

    dim3 grid(NT_, H_, B_);   // 13 x 10 x 64 = 8320 workgroups
    attn_super_kernel<<<grid, 128, LDS_FLOATS * sizeof(float), stream>>>(
        x, ktv, kth, vtv, vth, out);
}